// CenterNetDecoder_24068996726834
// MI455X (gfx1250) — compile-verified
//
#include <hip/hip_runtime.h>
#include <math.h>

#define HM_H 512
#define HM_W 512
#define HWSZ (HM_H*HM_W)
#define BATCH 16
#define NCH 2
#define KSEL 100
#define NPT 200          // NCH*KSEL
#define NBINS 2048
#define SN 2048          // candidate/sort buffer size (pow2)
#define JT 13            // ceil(200/16)
#define NIT 256          // row-pair iterations per channel
#define ROWF 512
#define STAGEF (6*ROWF)  // 4 own rows + 2 other rows per stage buffer

typedef __attribute__((ext_vector_type(2))) float v2f;
typedef __attribute__((ext_vector_type(8))) float v8f;

// monotonic float->uint key (larger float => larger key)
__device__ __forceinline__ unsigned fkey(float f) {
    unsigned u = __float_as_uint(f);
    return (u & 0x80000000u) ? ~u : (u | 0x80000000u);
}

// Issue async DMA of the 6-row stage tile for iteration `iter` into `buf`.
// 768 lanes each move one 16B chunk: slots 0..3 = own rows r0-1..r0+2,
// slots 4..5 = other rows r0..r0+1. Out-of-range rows are EXEC-masked off
// (their LDS slots are never read).
__device__ __forceinline__ void async_stage(const float* __restrict__ own,
                                            const float* __restrict__ other,
                                            float* buf, int iter, int tid) {
    if (tid < 768) {
        int slot = tid >> 7;            // 0..5
        int colb = (tid & 127) << 4;    // byte offset within row
        int r0 = iter * 2;
        int gr; bool okrow; const float* src;
        if (slot < 4) { gr = r0 - 1 + slot; okrow = (gr >= 0) && (gr < HM_H); src = own; }
        else          { gr = r0 + (slot - 4); okrow = true; src = other; }
        if (okrow) {
            const char* g = (const char*)(src + gr * HM_W) + colb;
            unsigned lds = (unsigned)(uintptr_t)((char*)buf + (slot << 11) + colb);
            asm volatile("global_load_async_to_lds_b128 %0, %1, off"
                         :: "v"(lds), "v"((unsigned long long)(uintptr_t)g)
                         : "memory");
        }
    }
}

// Double-buffered async-staged scan of one channel: calls fn(pixel_idx, hm_value).
template <typename F>
__device__ __forceinline__ void scan_channel(const float* __restrict__ own,
                                             const float* __restrict__ other,
                                             float (&bufs)[2][STAGEF],
                                             int tid, F fn) {
    async_stage(own, other, bufs[0], 0, tid);
    for (int i = 0; i < NIT; ++i) {
        if (i + 1 < NIT) {
            async_stage(own, other, bufs[(i + 1) & 1], i + 1, tid);
            asm volatile("s_wait_asynccnt 0x1" ::: "memory");  // current tile done
        } else {
            asm volatile("s_wait_asynccnt 0x0" ::: "memory");
        }
        __syncthreads();
        const float* buf  = bufs[i & 1];
        const float* ownb = buf;
        const float* othb = buf + 4 * ROWF;
        int lr = tid >> 9;              // 0,1: which of the two center rows
        int x  = tid & (HM_W - 1);
        int y  = i * 2 + lr;
        float m = ownb[(lr + 1) * ROWF + x];
        float o = othb[lr * ROWF + x];
        float pooled = m;
#pragma unroll
        for (int dy = -1; dy <= 1; ++dy) {
            int yy = y + dy;
            if (yy < 0 || yy >= HM_H) continue;
            int slot = lr + 1 + dy;
#pragma unroll
            for (int dx = -1; dx <= 1; ++dx) {
                int xx = x + dx;
                if (xx < 0 || xx >= HM_W) continue;
                pooled = fmaxf(pooled, ownb[slot * ROWF + xx]);
            }
        }
        float val = (m >= o) ? pooled : 0.0f;
        fn(i * 1024 + tid, val);        // pixel idx == y*512 + x
        __syncthreads();                // protect buffer reuse next iteration
    }
}

// One workgroup per (heatmap, batch, channel): exact top-100 via
// histogram threshold select + bitonic sort in LDS.
__global__ __launch_bounds__(1024)
void topk_select_kernel(const float* __restrict__ tl,
                        const float* __restrict__ br,
                        const float* __restrict__ ct,
                        float* __restrict__ SC,
                        float* __restrict__ XS,
                        float* __restrict__ YS) {
    __shared__ __align__(16) float bufs[2][STAGEF];
    __shared__ unsigned hist[NBINS];
    __shared__ float ts[SN];
    __shared__ unsigned ti[SN];
    __shared__ int sh_tbin;
    __shared__ unsigned cnt;

    const int gid = blockIdx.x;        // 0..95
    const int hm  = gid >> 5;          // which heatmap
    const int rem = gid & 31;
    const int b   = rem >> 1;
    const int c   = rem & 1;
    const float* heat  = (hm == 0) ? tl : ((hm == 1) ? br : ct);
    const float* own   = heat + (size_t)(b * NCH + c) * HWSZ;
    const float* other = heat + (size_t)(b * NCH + (1 - c)) * HWSZ;
    const int tid = threadIdx.x;

    for (int i = tid; i < NBINS; i += 1024) hist[i] = 0;
    __syncthreads();

    // pass 1: histogram of keys (top 11 bits -> 2048 bins)
    scan_channel(own, other, bufs, tid, [&](int idx, float val) {
        (void)idx;
        atomicAdd(&hist[fkey(val) >> 21], 1u);
    });
    __syncthreads();

    // find threshold bin: first bin (from top) where cumulative count >= K
    if (tid == 0) {
        int cum = 0, t = 0;
        for (int bin = NBINS - 1; bin >= 0; --bin) {
            int h = (int)hist[bin];
            if (cum + h >= KSEL) { t = bin; break; }
            cum += h;
        }
        sh_tbin = t;
        cnt = 0;
    }
    __syncthreads();
    const unsigned tbin = (unsigned)sh_tbin;

    for (int i = tid; i < SN; i += 1024) { ts[i] = -INFINITY; ti[i] = 0; }
    __syncthreads();

    // pass 2: collect all candidates with bin >= threshold bin
    scan_channel(own, other, bufs, tid, [&](int idx, float val) {
        if ((fkey(val) >> 21) >= tbin) {
            unsigned p = atomicAdd(&cnt, 1u);
            if (p < SN) { ts[p] = val; ti[p] = (unsigned)idx; }
        }
    });
    __syncthreads();

    // bitonic sort descending over SN elements (1024 threads, 2 elems/thread)
    for (unsigned kk = 2; kk <= SN; kk <<= 1) {
        for (unsigned jj = kk >> 1; jj > 0; jj >>= 1) {
            for (unsigned i = tid; i < SN; i += 1024) {
                unsigned ixj = i ^ jj;
                if (ixj > i) {
                    bool desc = ((i & kk) == 0);
                    float a = ts[i], bb2 = ts[ixj];
                    bool sw = desc ? (a < bb2) : (a > bb2);
                    if (sw) {
                        ts[i] = bb2; ts[ixj] = a;
                        unsigned tt = ti[i]; ti[i] = ti[ixj]; ti[ixj] = tt;
                    }
                }
            }
            __syncthreads();
        }
    }

    // emit top-100 (sorted descending): scores, xs, ys
    if (tid < KSEL) {
        int n = c * KSEL + tid;
        int base = (hm * BATCH + b) * NPT + n;
        unsigned idx = ti[tid];
        SC[base] = ts[tid];
        XS[base] = (float)(idx & (HM_W - 1));
        YS[base] = (float)(idx >> 9);
    }
}

// One wave per (batch, tl index i, 16-wide br tile). The center-distance
// test d2 = (cx-ctx)^2+(cy-cty)^2 is expanded to a K=4 GEMM:
//   A row m  = [cx_m, cy_m, 1, 0]          (16 pairs, fixed i, 16 j's)
//   B col n  = [-2ctx_n, -2cty_n, ctx_n^2+cty_n^2, 0]
//   d2[m][n] = D[m][n] + (cx_m^2+cy_m^2)
// computed with V_WMMA_F32_16X16X4_F32, 13 N-tiles over 200 centers.
__global__ __launch_bounds__(32)
void pair_decode_kernel(const float* __restrict__ SC,
                        const float* __restrict__ XS,
                        const float* __restrict__ YS,
                        const float* __restrict__ bbox,
                        const int* __restrict__ imh,
                        const int* __restrict__ imw,
                        float* __restrict__ out) {
    const int bid = blockIdx.x;
    const int b  = bid / (NPT * JT);
    const int r  = bid % (NPT * JT);
    const int i  = r / JT;
    const int jt = r % JT;
    const int lane = threadIdx.x;
    const int lj = lane & 15;
    const int j = jt * 16 + lj;
    const bool lowh = (lane < 16);

    // tl point (broadcast), br point per row
    float tl_x = XS[(0 * BATCH + b) * NPT + i];
    float tl_y = YS[(0 * BATCH + b) * NPT + i];
    float tl_s = SC[(0 * BATCH + b) * NPT + i];
    bool jok = (j < NPT);
    float br_x = jok ? XS[(1 * BATCH + b) * NPT + j] : -1e30f;
    float br_y = jok ? YS[(1 * BATCH + b) * NPT + j] : -1e30f;
    float br_s = jok ? SC[(1 * BATCH + b) * NPT + j] : 0.0f;

    float cx = 0.5f * (tl_x + br_x);
    float cy = 0.5f * (tl_y + br_y);
    float rowterm = cx * cx + cy * cy;     // lives in lane m for row m

    // A matrix, 16x4 f32 layout: lanes 0-15 hold K=0,1; lanes 16-31 hold K=2,3
    v2f a;
    a.x = lowh ? cx : 1.0f;
    a.y = lowh ? cy : 0.0f;

    const float* ctX = XS + (2 * BATCH + b) * NPT;
    const float* ctY = YS + (2 * BATCH + b) * NPT;

    unsigned bits = 0;   // bit r == (exists center with d2<4) for row (r, r+8)
    for (int tt = 0; tt < JT; ++tt) {
        int n = tt * 16 + lj;
        bool nok = (n < NPT);
        float ctx = nok ? ctX[n] : 1e18f;   // pad -> d2 = +inf, never "near"
        float cty = nok ? ctY[n] : 1e18f;
        // B matrix, 4x16 f32 layout: lanes 0-15 hold K=0,1 rows; 16-31 hold K=2,3
        v2f bv;
        bv.x = lowh ? (-2.0f * ctx) : (ctx * ctx + cty * cty);
        bv.y = lowh ? (-2.0f * cty) : 0.0f;
        v8f acc = {};
        acc = __builtin_amdgcn_wmma_f32_16x16x4_f32(
            /*neg_a=*/false, a, /*neg_b=*/false, bv,
            /*c_mod=*/(short)0, acc, /*reuse_a=*/false, /*reuse_b=*/false);
#pragma unroll
        for (int rr = 0; rr < 8; ++rr) {
            int m = rr + (lowh ? 0 : 8);               // row of this VGPR slot
            float rt = __shfl(rowterm, m, 32);
            if (acc[rr] + rt < 4.0f) bits |= (1u << rr);
        }
    }
    // OR-reduce across the 16 columns in each lane half
#pragma unroll
    for (int off = 1; off < 16; off <<= 1) bits |= __shfl_xor(bits, off, 16);
    unsigned lowv = __shfl(bits, 0, 32);   // rows 0..7
    unsigned hiv  = __shfl(bits, 16, 32);  // rows 8..15

    if (lowh && jok) {
        bool near = (lane < 8) ? (((lowv >> lane) & 1u) != 0u)
                               : (((hiv >> (lane - 8)) & 1u) != 0u);
        bool valid = (br_x > tl_x) && (br_y > tl_y);
        float score = 0.5f * (tl_s + br_s);
        float kf = (valid && near && (score >= 0.1f)) ? 1.0f : 0.0f;
        float cls = (float)(i / KSEL);
        float sx = (float)imw[0] / (float)HM_W;
        float sy = (float)imh[0] / (float)HM_H;

        // corner boxes: (B, 2, N, N, 6), slot 0
        size_t base0 = ((((size_t)b * 2 + 0) * NPT + i) * NPT + j) * 6;
        float2* o0 = (float2*)(out + base0);
        o0[0] = make_float2(tl_x * sx * kf, tl_y * sy * kf);
        o0[1] = make_float2(br_x * sx * kf, br_y * sy * kf);
        o0[2] = make_float2(score * kf, cls * kf);

        // regression boxes: gather bbox at clipped center pixel
        int cxi = min(max((int)cx, 0), HM_W - 1);
        int cyi = min(max((int)cy, 0), HM_H - 1);
        int flat = cyi * HM_W + cxi;
        const float* bb0 = bbox + (size_t)b * 4 * HWSZ;
        float pcx = bb0[flat];
        float pcy = bb0[HWSZ + flat];
        float w   = bb0[2 * HWSZ + flat];
        float h   = bb0[3 * HWSZ + flat];
        size_t base1 = ((((size_t)b * 2 + 1) * NPT + i) * NPT + j) * 6;
        float2* o1 = (float2*)(out + base1);
        o1[0] = make_float2((pcx - 0.5f * w) * sx * kf, (pcy - 0.5f * h) * sy * kf);
        o1[1] = make_float2((pcx + 0.5f * w) * sx * kf, (pcy + 0.5f * h) * sy * kf);
        o1[2] = make_float2(score * kf, cls * kf);
    }
}

extern "C" void kernel_launch(void* const* d_in, const int* in_sizes, int n_in,
                              void* d_out, int out_size, void* d_ws, size_t ws_size,
                              hipStream_t stream) {
    const float* tl   = (const float*)d_in[0];
    const float* br   = (const float*)d_in[1];
    const float* ct   = (const float*)d_in[2];
    const float* bbox = (const float*)d_in[3];
    const int*   imh  = (const int*)d_in[4];
    const int*   imw  = (const int*)d_in[5];

    float* ws = (float*)d_ws;            // 3*16*200 floats per array (~115 KB total)
    float* SC = ws;
    float* XS = ws + 3 * BATCH * NPT;
    float* YS = ws + 6 * BATCH * NPT;

    hipLaunchKernelGGL(topk_select_kernel, dim3(96), dim3(1024), 0, stream,
                       tl, br, ct, SC, XS, YS);
    hipLaunchKernelGGL(pair_decode_kernel, dim3(BATCH * NPT * JT), dim3(32), 0, stream,
                       SC, XS, YS, bbox, imh, imw, (float*)d_out);
}